// vectorGraph_8358006358517
// MI455X (gfx1250) — compile-verified
//
#include <hip/hip_runtime.h>
#include <hip/hip_bf16.h>

#ifndef __has_builtin
#define __has_builtin(x) 0
#endif

#define EPSF 1e-5f

// ---- CDNA5 async-to-LDS path (guarded; falls back to plain loads) ----------
#if __has_builtin(__builtin_amdgcn_global_load_async_to_lds_b128)
#define HAVE_ASYNC_B128 1
#else
#define HAVE_ASYNC_B128 0
#endif

#if __has_builtin(__builtin_amdgcn_s_wait_asynccnt)
#define WAIT_ASYNC0() __builtin_amdgcn_s_wait_asynccnt(0)
#else
#define WAIT_ASYNC0() asm volatile("s_wait_asynccnt 0" ::: "memory")
#endif

// Builtin parameter types (from hipcc diagnostic):
//   param0: __attribute__((address_space(1))) int4*  (global source)
//   param1: __attribute__((address_space(3))) int4*  (LDS destination)
typedef int v4i __attribute__((ext_vector_type(4)));
typedef __attribute__((address_space(1))) v4i as1_v4i;
typedef __attribute__((address_space(3))) v4i as3_v4i;

// Native L2-side fp32 atomic add (global_atomic_add_f32), not a CAS loop.
__device__ __forceinline__ void atomAddF(float* p, float v) {
    (void)unsafeAtomicAdd(p, v);
}

// ---------------------------------------------------------------------------
// Shared per-edge math + 12 scatter atomics.
// a[9], b[9] = node columns for endpoints i, j of one channel.
// lp/p1/p2 already offset to channel base (c*3*N).
// ---------------------------------------------------------------------------
__device__ __forceinline__ void edge_body(const float a[9], const float b[9],
                                          float* __restrict__ lp,
                                          float* __restrict__ p1,
                                          float* __restrict__ p2,
                                          int i, int j, int N) {
    float gx = a[0] - b[0], gy = a[1] - b[1], gz = a[2] - b[2];

    float ux = a[0] - a[3], uy = a[1] - a[4], uz = a[2] - a[5];
    float wx = a[0] - a[6], wy = a[1] - a[7], wz = a[2] - a[8];
    float n1x = uy * wz - uz * wy;
    float n1y = uz * wx - ux * wz;
    float n1z = ux * wy - uy * wx;

    float px = b[0] - b[3], py = b[1] - b[4], pz = b[2] - b[5];
    float qx = b[0] - b[6], qy = b[1] - b[7], qz = b[2] - b[8];
    float n2x = py * qz - pz * qy;
    float n2y = pz * qx - px * qz;
    float n2z = px * qy - py * qx;

    float r1 = rsqrtf(n1x * n1x + n1y * n1y + n1z * n1z + EPSF);
    float r2 = rsqrtf(n2x * n2x + n2y * n2y + n2z * n2z + EPSF);
    n1x *= r1; n1y *= r1; n1z *= r1;
    n2x *= r2; n2y *= r2; n2z *= r2;

    float vx = n1y * n2z - n1z * n2y;
    float vy = n1z * n2x - n1x * n2z;
    float vz = n1x * n2y - n1y * n2x;
    float ang = vx * vx + vy * vy + vz * vz;

    float wgx = ang * gx, wgy = ang * gy, wgz = ang * gz;

    atomAddF(lp + (size_t)0 * N + i,  wgx);
    atomAddF(lp + (size_t)1 * N + i,  wgy);
    atomAddF(lp + (size_t)2 * N + i,  wgz);
    atomAddF(lp + (size_t)0 * N + j, -wgx);
    atomAddF(lp + (size_t)1 * N + j, -wgy);
    atomAddF(lp + (size_t)2 * N + j, -wgz);

    atomAddF(p1 + (size_t)0 * N + i, gx);
    atomAddF(p1 + (size_t)1 * N + i, gy);
    atomAddF(p1 + (size_t)2 * N + i, gz);

    atomAddF(p2 + (size_t)0 * N + j, gx);
    atomAddF(p2 + (size_t)1 * N + j, gy);
    atomAddF(p2 + (size_t)2 * N + j, gz);
}

// ---------------------------------------------------------------------------
// Zero the 3 accumulator arrays (3*M floats), float4-wide.
// ---------------------------------------------------------------------------
__global__ __launch_bounds__(256) void vg_zero_kernel(float* __restrict__ w, long long n) {
    const long long stride = (long long)gridDim.x * blockDim.x;
    const long long n4 = n >> 2;
    float4* __restrict__ w4 = (float4*)w;
    const float4 z = make_float4(0.f, 0.f, 0.f, 0.f);
    for (long long t = (long long)blockIdx.x * blockDim.x + threadIdx.x; t < n4; t += stride)
        w4[t] = z;
    for (long long t = (n4 << 2) + (long long)blockIdx.x * blockDim.x + threadIdx.x; t < n; t += stride)
        w[t] = 0.f;
}

// ---------------------------------------------------------------------------
// Transpose x (C,9,N) -> three float4-per-node arrays so the edge gather is
// 6 x b128 requests instead of 18 x b32.  Reads are row-coalesced.
// ---------------------------------------------------------------------------
__global__ __launch_bounds__(256) void vg_transpose_kernel(
    const float* __restrict__ x, const int* __restrict__ nnodes,
    float4* __restrict__ t0, float4* __restrict__ t1, float4* __restrict__ t2,
    long long CN) {
    const int N = nnodes[0];
    const long long stride = (long long)gridDim.x * blockDim.x;
    for (long long w = (long long)blockIdx.x * blockDim.x + threadIdx.x; w < CN; w += stride) {
        const int c = (int)(w / N);
        const long long n = w - (long long)c * N;
        const float* __restrict__ xr = x + (size_t)c * 9 * N + n;
        t0[w] = make_float4(xr[0],             xr[(size_t)N],     xr[(size_t)2 * N], 0.f);
        t1[w] = make_float4(xr[(size_t)3 * N], xr[(size_t)4 * N], xr[(size_t)5 * N], 0.f);
        t2[w] = make_float4(xr[(size_t)6 * N], xr[(size_t)7 * N], xr[(size_t)8 * N], 0.f);
    }
}

// ---------------------------------------------------------------------------
// Edge kernel, transposed-gather variant: 6 x global_load_b128 per work item.
// One (edge, channel) per lane; t = c*E + e so index loads are coalesced.
// ---------------------------------------------------------------------------
__global__ __launch_bounds__(256) void vg_edge_t_kernel(
    const float4* __restrict__ t0, const float4* __restrict__ t1,
    const float4* __restrict__ t2,
    const int* __restrict__ iInd, const int* __restrict__ jInd,
    const int* __restrict__ nnodes,
    float* __restrict__ lap, float* __restrict__ acc1, float* __restrict__ acc2,
    int E, int M) {
    const int N  = nnodes[0];
    const int CH = M / (3 * N);
    const long long total  = (long long)E * (long long)CH;
    const long long stride = (long long)gridDim.x * blockDim.x;

    for (long long t = (long long)blockIdx.x * blockDim.x + threadIdx.x; t < total; t += stride) {
        const int e = (int)(t % (long long)E);
        const int c = (int)(t / (long long)E);

        // Run-ahead prefetch of the streaming index arrays (global_prefetch_b8).
        if (((e & 63) == 0) && (e + 4096 < E)) {
            __builtin_prefetch(iInd + e + 4096, 0, 1);
            __builtin_prefetch(jInd + e + 4096, 0, 1);
        }

        const int i = iInd[e];
        const int j = jInd[e];

        const size_t base = (size_t)c * N;
        const float4 A0 = t0[base + i], A1 = t1[base + i], A2 = t2[base + i];
        const float4 B0 = t0[base + j], B1 = t1[base + j], B2 = t2[base + j];

        const float a[9] = {A0.x, A0.y, A0.z, A1.x, A1.y, A1.z, A2.x, A2.y, A2.z};
        const float b[9] = {B0.x, B0.y, B0.z, B1.x, B1.y, B1.z, B2.x, B2.y, B2.z};

        float* __restrict__ lp = lap  + (size_t)c * 3 * N;
        float* __restrict__ p1 = acc1 + (size_t)c * 3 * N;
        float* __restrict__ p2 = acc2 + (size_t)c * 3 * N;
        edge_body(a, b, lp, p1, p2, i, j, N);
    }
}

// ---------------------------------------------------------------------------
// Edge kernel, direct-gather fallback (used if scratch is too small for the
// transposed copies): 18 scattered b32 loads per work item.
// ---------------------------------------------------------------------------
__global__ __launch_bounds__(256) void vg_edge_kernel(
    const float* __restrict__ x,
    const int* __restrict__ iInd, const int* __restrict__ jInd,
    const int* __restrict__ nnodes,
    float* __restrict__ lap, float* __restrict__ acc1, float* __restrict__ acc2,
    int E, int M) {
    const int N  = nnodes[0];
    const int CH = M / (3 * N);
    const long long total  = (long long)E * (long long)CH;
    const long long stride = (long long)gridDim.x * blockDim.x;

    for (long long t = (long long)blockIdx.x * blockDim.x + threadIdx.x; t < total; t += stride) {
        const int e = (int)(t % (long long)E);
        const int c = (int)(t / (long long)E);

        if (((e & 63) == 0) && (e + 4096 < E)) {
            __builtin_prefetch(iInd + e + 4096, 0, 1);
            __builtin_prefetch(jInd + e + 4096, 0, 1);
        }

        const int i = iInd[e];
        const int j = jInd[e];

        const float* __restrict__ xa = x + (size_t)c * 9 * N;
        float a[9], b[9];
#pragma unroll
        for (int k = 0; k < 9; ++k) a[k] = xa[(size_t)k * N + i];
#pragma unroll
        for (int k = 0; k < 9; ++k) b[k] = xa[(size_t)k * N + j];

        float* __restrict__ lp = lap  + (size_t)c * 3 * N;
        float* __restrict__ p1 = acc1 + (size_t)c * 3 * N;
        float* __restrict__ p2 = acc2 + (size_t)c * 3 * N;
        edge_body(a, b, lp, p1, p2, i, j, N);
    }
}

// ---------------------------------------------------------------------------
// Finalize: out = lap + max(x1, x2).  Streaming pass; float4-wide.  The three
// input streams are staged through LDS with CDNA5 async-to-LDS DMA (b128),
// overlapping VMEM with the DS reads; plain b128 otherwise.
// ---------------------------------------------------------------------------
__global__ __launch_bounds__(256) void vg_final_kernel(
    const float* __restrict__ lap, const float* __restrict__ a1,
    const float* __restrict__ a2, float* __restrict__ out, int M) {
    const long long stride = (long long)gridDim.x * blockDim.x;

    if ((M & 3) == 0) {
        const long long M4 = (long long)M >> 2;
        const float4* __restrict__ l4 = (const float4*)lap;
        const float4* __restrict__ x4 = (const float4*)a1;
        const float4* __restrict__ y4 = (const float4*)a2;
        float4* __restrict__ o4 = (float4*)out;

#if HAVE_ASYNC_B128
        __shared__ float4 sL[256];
        __shared__ float4 sX[256];
        __shared__ float4 sY[256];
#endif
        for (long long t = (long long)blockIdx.x * blockDim.x + threadIdx.x; t < M4; t += stride) {
#if HAVE_ASYNC_B128
            __builtin_amdgcn_global_load_async_to_lds_b128(
                (as1_v4i*)const_cast<float4*>(l4 + t), (as3_v4i*)&sL[threadIdx.x], 0, 0);
            __builtin_amdgcn_global_load_async_to_lds_b128(
                (as1_v4i*)const_cast<float4*>(x4 + t), (as3_v4i*)&sX[threadIdx.x], 0, 0);
            __builtin_amdgcn_global_load_async_to_lds_b128(
                (as1_v4i*)const_cast<float4*>(y4 + t), (as3_v4i*)&sY[threadIdx.x], 0, 0);
            WAIT_ASYNC0();
            const float4 L = sL[threadIdx.x];
            const float4 X = sX[threadIdx.x];
            const float4 Y = sY[threadIdx.x];
#else
            const float4 L = l4[t];
            const float4 X = x4[t];
            const float4 Y = y4[t];
#endif
            float4 o;
            o.x = L.x + fmaxf(X.x, Y.x);
            o.y = L.y + fmaxf(X.y, Y.y);
            o.z = L.z + fmaxf(X.z, Y.z);
            o.w = L.w + fmaxf(X.w, Y.w);
            o4[t] = o;
        }
    } else {
        for (long long t = (long long)blockIdx.x * blockDim.x + threadIdx.x; t < M; t += stride)
            out[t] = lap[t] + fmaxf(a1[t], a2[t]);
    }
}

// ---------------------------------------------------------------------------
// Host launcher.  nnodes is a device scalar (cannot be read during graph
// capture), so kernels read N themselves; host grids are sized from E and M
// with grid-stride loops as the safety net.
// ---------------------------------------------------------------------------
extern "C" void kernel_launch(void* const* d_in, const int* in_sizes, int n_in,
                              void* d_out, int out_size, void* d_ws, size_t ws_size,
                              hipStream_t stream) {
    const float* x    = (const float*)d_in[0];
    const int*   iInd = (const int*)d_in[1];
    const int*   jInd = (const int*)d_in[2];
    const int*   nn   = (const int*)d_in[3];
    float*       out  = (float*)d_out;

    const int E = in_sizes[1];
    const int M = out_size;        // B*C*3*N
    const long long CN = (long long)M / 3;  // B*C*N

    float* lap  = (float*)d_ws;
    float* acc1 = lap + (size_t)M;
    float* acc2 = lap + (size_t)2 * M;

    const size_t accBytes = (size_t)3 * (size_t)M * sizeof(float);
    const size_t tOff     = (accBytes + 15) & ~(size_t)15;
    const size_t tBytes   = (size_t)3 * (size_t)CN * sizeof(float4);
    const bool   useT     = (ws_size >= tOff + tBytes) && (M % 3 == 0);

    float4* t0 = (float4*)((char*)d_ws + tOff);
    float4* t1 = t0 + (size_t)CN;
    float4* t2 = t1 + (size_t)CN;

    // 1) zero the accumulators
    {
        const long long n = 3LL * M;
        long long blocks = (n / 4 + 255) / 256;
        if (blocks < 1) blocks = 1;
        if (blocks > (1 << 20)) blocks = (1 << 20);
        vg_zero_kernel<<<(unsigned)blocks, 256, 0, stream>>>((float*)d_ws, n);
    }

    // Host-side guess of channel count (N=50000 in this harness); grid-stride
    // loops in the kernels make correctness independent of the guess.
    int chGuess = (M % 150000 == 0) ? (M / 150000) : 8;
    if (chGuess < 1) chGuess = 1;
    const long long totWork = (long long)E * chGuess;
    long long eBlocks = (totWork + 255) / 256;
    if (eBlocks < 1) eBlocks = 1;
    if (eBlocks > (1 << 18)) eBlocks = (1 << 18);

    // 2) edge gather/compute/scatter
    if (useT) {
        long long tBlocks = (CN + 255) / 256;
        if (tBlocks < 1) tBlocks = 1;
        if (tBlocks > (1 << 18)) tBlocks = (1 << 18);
        vg_transpose_kernel<<<(unsigned)tBlocks, 256, 0, stream>>>(x, nn, t0, t1, t2, CN);
        vg_edge_t_kernel<<<(unsigned)eBlocks, 256, 0, stream>>>(
            t0, t1, t2, iInd, jInd, nn, lap, acc1, acc2, E, M);
    } else {
        vg_edge_kernel<<<(unsigned)eBlocks, 256, 0, stream>>>(
            x, iInd, jInd, nn, lap, acc1, acc2, E, M);
    }

    // 3) finalize: out = lap + max(x1, x2)
    {
        long long fBlocks = ((long long)M / 4 + 255) / 256;
        if (fBlocks < 1) fBlocks = 1;
        if (fBlocks > (1 << 18)) fBlocks = (1 << 18);
        vg_final_kernel<<<(unsigned)fBlocks, 256, 0, stream>>>(lap, acc1, acc2, out, M);
    }
}